// QuantizationLayerEventCount_55078660604232
// MI455X (gfx1250) — compile-verified
//
#include <hip/hip_runtime.h>

// Sensor geometry from the reference
#define SENSOR_W 640
#define SENSOR_H 480

// One event = (x, y, t, p) as 4 packed floats = 16 bytes, 16B-aligned.
typedef float evf4 __attribute__((ext_vector_type(4)));

// ---------------------------------------------------------------------------
// Kernel 1: zero the 2*H*W float output (harness poisons it with 0xAA).
// Vector b128 stores; scalar tail for robustness.
// ---------------------------------------------------------------------------
__global__ __launch_bounds__(256) void qec_zero(float* __restrict__ out, int n)
{
    int i = (blockIdx.x * 256 + threadIdx.x) * 4;
    if (i + 3 < n) {
        *(evf4*)(out + i) = (evf4)0.0f;      // global_store_b128
    } else {
        for (int k = i; k < n; ++k) out[k] = 0.0f;
    }
}

// ---------------------------------------------------------------------------
// Flat voxel index, matching the reference exactly:
//   idx = x + W*y + W*H*((p+1)/2)
// x in [0,640), y in [0,480), p in {-1,+1}; all are exact whole-number floats
// (well below 2^24), so int conversion is exact.
// ---------------------------------------------------------------------------
__device__ __forceinline__ int qec_idx(evf4 e)
{
    int ip = (int)((e.w + 1.0f) * 0.5f);               // {-1,+1} -> {0,1}
    return (int)e.x + SENSOR_W * (int)e.y + (SENSOR_W * SENSOR_H) * ip;
}

// ---------------------------------------------------------------------------
// Kernel 2: scatter. Because the reference binarizes (vox>0 -> 1), every
// event simply stores 1.0f to its voxel — all racing writers store the same
// value, so NO atomics are required.
//
// Memory behavior (the thing MI455X rewards):
//   * events are a 268 MB use-once stream  -> non-temporal b128 loads so we
//     don't churn the 192 MB L2 and evict the hot 2.4 MB output image
//   * output (2.4 MB) stays L2-resident    -> regular-temporal b32 stores
//   * 4-way unroll: 4 independent global_load_b128 in flight per thread
//     before the dependent index math / stores
// ---------------------------------------------------------------------------
__global__ __launch_bounds__(256) void qec_scatter(const evf4* __restrict__ ev,
                                                   float* __restrict__ out,
                                                   int n)
{
    const int stride = gridDim.x * 256;
    int i = blockIdx.x * 256 + threadIdx.x;

    // main unrolled-by-4 grid-stride loop (lane-contiguous => fully coalesced)
    for (; i + 3 * stride < n; i += 4 * stride) {
        evf4 e0 = __builtin_nontemporal_load(ev + i);
        evf4 e1 = __builtin_nontemporal_load(ev + i + stride);
        evf4 e2 = __builtin_nontemporal_load(ev + i + 2 * stride);
        evf4 e3 = __builtin_nontemporal_load(ev + i + 3 * stride);
        // speculative prefetch of the next chunk (global_prefetch_b8);
        // translation failures past the end are silently dropped
        __builtin_prefetch(ev + i + 4 * stride, 0, 3);
        out[qec_idx(e0)] = 1.0f;
        out[qec_idx(e1)] = 1.0f;
        out[qec_idx(e2)] = 1.0f;
        out[qec_idx(e3)] = 1.0f;
    }
    // tail
    for (; i < n; i += stride) {
        evf4 e = __builtin_nontemporal_load(ev + i);
        out[qec_idx(e)] = 1.0f;
    }
}

// ---------------------------------------------------------------------------
// Host-side launcher. Deterministic; everything on `stream`; no APIs that
// break graph capture.
// ---------------------------------------------------------------------------
extern "C" void kernel_launch(void* const* d_in, const int* in_sizes, int n_in,
                              void* d_out, int out_size, void* d_ws, size_t ws_size,
                              hipStream_t stream)
{
    (void)n_in; (void)d_ws; (void)ws_size;

    const evf4* ev  = (const evf4*)d_in[0];
    float*      out = (float*)d_out;

    const int n_events = in_sizes[0] / 4;   // events array is [N,4] floats

    // 1) zero the output (out_size = 2*480*640 = 614400 floats)
    int gz = (out_size + 4 * 256 - 1) / (4 * 256);
    if (gz < 1) gz = 1;
    qec_zero<<<gz, 256, 0, stream>>>(out, out_size);

    // 2) scatter: cap at 8192 blocks (2M threads -> ~8 events/thread at N=16.7M),
    //    grid-stride loop covers any N
    int gs = (n_events + 255) / 256;
    if (gs > 8192) gs = 8192;
    if (gs < 1)    gs = 1;
    qec_scatter<<<gs, 256, 0, stream>>>(ev, out, n_events);
}